// NNModel2_58153857188028
// MI455X (gfx1250) — compile-verified
//
#include <hip/hip_runtime.h>

#define NPTS 20000
#define NLVL 7
#define NN   2048
#define DIM  3
#define H    16
#define CB   64
#define HTOT (NLVL*H)   // 112
#define KEPS 1e-12f

typedef float v2f __attribute__((ext_vector_type(2)));
typedef float v8f __attribute__((ext_vector_type(8)));

// D = A(16x4) * B(4x16) + C, full fp32 (V_WMMA_F32_16X16X4_F32)
__device__ __forceinline__ v8f wmma4(v2f a, v2f b, v8f c) {
    return __builtin_amdgcn_wmma_f32_16x16x4_f32(false, a, false, b,
                                                 (short)0, c, false, false);
}

// ---------------------------------------------------------------- build K_xx
__global__ void build_kxx(float* __restrict__ A, const float* __restrict__ tp,
                          const float* __restrict__ sigma,
                          const float* __restrict__ alpha, int l) {
    int idx = blockIdx.x * blockDim.x + threadIdx.x;
    if (idx >= NN * NN) return;
    int i = idx / NN, j = idx % NN;
    const float* xt = tp + (size_t)l * NN * DIM;
    float dx = xt[i*3+0] - xt[j*3+0];
    float dy = xt[i*3+1] - xt[j*3+1];
    float dz = xt[i*3+2] - xt[j*3+2];
    float d2 = dx*dx + dy*dy + dz*dz;
    float v = __expf(-sqrtf(fmaxf(d2, KEPS)) / sigma[l]);
    if (i == j) v += alpha[l];
    A[idx] = v;
}

// --------------------------------------------------- Cholesky diagonal block
__global__ void chol_potrf(float* __restrict__ A, int k) {
    __shared__ float T[CB][CB + 1];
    int tid = threadIdx.x;          // 64 threads
    for (int r = 0; r < CB; ++r) T[r][tid] = A[(size_t)(k + r) * NN + k + tid];
    __syncthreads();
    for (int c = 0; c < CB; ++c) {
        if (tid == c) T[c][c] = sqrtf(T[c][c]);
        __syncthreads();
        if (tid > c) T[tid][c] /= T[c][c];
        __syncthreads();
        if (tid > c) {
            float lc = T[tid][c];
            for (int m = c + 1; m <= tid; ++m) T[tid][m] -= lc * T[m][c];
        }
        __syncthreads();
    }
    for (int r = 0; r < CB; ++r) A[(size_t)(k + r) * NN + k + tid] = T[r][tid];
}

// ------------------------------------------ panel TRSM: L21 = A21 * L11^{-T}
__global__ void chol_trsm(float* __restrict__ A, int k) {
    __shared__ float L[CB][CB + 1];
    int tid = threadIdx.x;          // 64 threads
    for (int r = 0; r < CB; ++r) L[r][tid] = A[(size_t)(k + r) * NN + k + tid];
    __syncthreads();
    int row = k + CB + blockIdx.x * CB + tid;
    float v[CB];
#pragma unroll
    for (int c = 0; c < CB; ++c) v[c] = A[(size_t)row * NN + k + c];
#pragma unroll
    for (int c = 0; c < CB; ++c) {
        float acc = v[c];
        for (int m = 0; m < c; ++m) acc -= v[m] * L[c][m];
        v[c] = acc / L[c][c];
    }
#pragma unroll
    for (int c = 0; c < CB; ++c) A[(size_t)row * NN + k + c] = v[c];
}

// ------------------------- trailing update (WMMA f32): C -= L21 * L21^T
// Pr is pre-negated at the LDS fill so the inner loop is pure DS->WMMA.
__global__ void chol_syrk(float* __restrict__ A, int k) {
    __shared__ float Pr[CB][CB + 2];   // negated row panel (A fragments)
    __shared__ float Pc[CB][CB + 2];   // col panel (B fragments)
    int T  = NN - k - CB;
    int nb = T / CB;
    int bi = blockIdx.x / nb, bj = blockIdx.x % nb;
    int rb = k + CB + bi * CB;
    int cbse = k + CB + bj * CB;
    int tid = threadIdx.x;          // 256 threads
    for (int t = tid; t < CB * CB; t += 256) {
        int r = t / CB, c = t % CB;
        Pr[r][c] = -A[(size_t)(rb + r) * NN + k + c];
        Pc[r][c] =  A[(size_t)(cbse + r) * NN + k + c];
    }
    __syncthreads();
    int wave = tid / 32, lane = tid % 32;
    int half = lane / 16, ln = lane % 16;
    for (int t = wave; t < 16; t += 8) {
        int ti = t >> 2, tj = t & 3;
        v8f c;
#pragma unroll
        for (int r = 0; r < 8; ++r)
            c[r] = A[(size_t)(rb + ti*16 + r + 8*half) * NN + cbse + tj*16 + ln];
#pragma unroll
        for (int kk = 0; kk < 16; ++kk) {
            int p = kk * 4 + 2 * half;
            v2f a = *(const v2f*)&Pr[ti*16 + ln][p];   // rows 8B-aligned (stride 66)
            v2f b = *(const v2f*)&Pc[tj*16 + ln][p];
            c = wmma4(a, b, c);
        }
#pragma unroll
        for (int r = 0; r < 8; ++r)
            A[(size_t)(rb + ti*16 + r + 8*half) * NN + cbse + tj*16 + ln] = c[r];
    }
}

// ---------------------------------------- forward substitution: L Y = W_emb
__global__ void solve_fwd(const float* __restrict__ A, const float* __restrict__ W,
                          float* __restrict__ Y, int l) {
    __shared__ float Lb[CB][CB + 1];
    __shared__ float S[CB][H];
    const float* Wl = W + (size_t)l * NN * H;
    int tid = threadIdx.x;          // 1024
    int r = tid / H, cc = tid % H;
    for (int blk = 0; blk < NN / CB; ++blk) {
        int row = blk * CB + r;
        const float* Arow = A + (size_t)row * NN;
        float a0 = 0.f, a1 = 0.f, a2 = 0.f, a3 = 0.f;
        int pe = blk * CB;
        for (int p = 0; p < pe; p += 4) {
            a0 += Arow[p+0] * Y[(p+0)*H + cc];
            a1 += Arow[p+1] * Y[(p+1)*H + cc];
            a2 += Arow[p+2] * Y[(p+2)*H + cc];
            a3 += Arow[p+3] * Y[(p+3)*H + cc];
        }
        float acc = Wl[row*H + cc] - (a0 + a1) - (a2 + a3);
        for (int t = tid; t < CB * CB; t += 1024)
            Lb[t/CB][t%CB] = A[(size_t)(blk*CB + t/CB) * NN + blk*CB + t%CB];
        S[r][cc] = acc;
        __syncthreads();
        if (tid < H) {
            int c = tid;
            for (int r2 = 0; r2 < CB; ++r2) {
                float v = S[r2][c];
                for (int m = 0; m < r2; ++m) v -= Lb[r2][m] * S[m][c];
                S[r2][c] = v / Lb[r2][r2];
            }
        }
        __syncthreads();
        Y[row*H + cc] = S[r][cc];
        __threadfence_block();
        __syncthreads();
    }
}

// --------------------------------------- backward substitution: L^T a = Y
__global__ void solve_bwd(const float* __restrict__ A, const float* __restrict__ Y,
                          float* __restrict__ X) {
    __shared__ float Lb[CB][CB + 1];
    __shared__ float S[CB][H];
    int tid = threadIdx.x;          // 1024
    int r = tid / H, cc = tid % H;
    for (int blk = NN / CB - 1; blk >= 0; --blk) {
        int row = blk * CB + r;
        float a0 = 0.f, a1 = 0.f, a2 = 0.f, a3 = 0.f;
        for (int p = (blk + 1) * CB; p < NN; p += 4) {
            a0 += A[(size_t)(p+0)*NN + row] * X[(p+0)*H + cc];
            a1 += A[(size_t)(p+1)*NN + row] * X[(p+1)*H + cc];
            a2 += A[(size_t)(p+2)*NN + row] * X[(p+2)*H + cc];
            a3 += A[(size_t)(p+3)*NN + row] * X[(p+3)*H + cc];
        }
        float acc = Y[row*H + cc] - (a0 + a1) - (a2 + a3);
        for (int t = tid; t < CB * CB; t += 1024)
            Lb[t/CB][t%CB] = A[(size_t)(blk*CB + t/CB) * NN + blk*CB + t%CB];
        S[r][cc] = acc;
        __syncthreads();
        if (tid < H) {
            int c = tid;
            for (int r2 = CB - 1; r2 >= 0; --r2) {
                float v = S[r2][c];
                for (int m = r2 + 1; m < CB; ++m) v -= Lb[m][r2] * S[m][c];
                S[r2][c] = v / Lb[r2][r2];
            }
        }
        __syncthreads();
        X[row*H + cc] = S[r][cc];
        __threadfence_block();
        __syncthreads();
    }
}

// ---------------- fused h_l = K(x, xt_l) @ a : exp generated into WMMA A-frag
__global__ void ktx_gemm(const float* __restrict__ x, const float* __restrict__ tp,
                         const float* __restrict__ sigma, const float* __restrict__ ac,
                         float* __restrict__ h, int l) {
    __shared__ float xt[CB][4];
    __shared__ float as[CB][H + 2];
    int tid = threadIdx.x;          // 256
    int wave = tid / 32, lane = tid % 32;
    int half = lane / 16, ln = lane % 16;
    int rowbase = blockIdx.x * 128 + wave * 16;
    int ar = rowbase + ln; if (ar >= NPTS) ar = NPTS - 1;
    float x0 = x[ar*3+0], x1 = x[ar*3+1], x2 = x[ar*3+2];
    float invs = 1.0f / sigma[l];
    const float* tpl = tp + (size_t)l * NN * DIM;
    v8f c = {0.f,0.f,0.f,0.f,0.f,0.f,0.f,0.f};
    for (int jb = 0; jb < NN; jb += CB) {
        for (int t = tid; t < CB * DIM; t += 256) xt[t/3][t%3] = tpl[jb*DIM + t];
        for (int t = tid; t < CB * H;   t += 256) as[t/H][t%H] = ac[(jb + t/H)*H + t%H];
        __syncthreads();
#pragma unroll
        for (int kk = 0; kk < 16; ++kk) {
            int p = kk * 4 + 2 * half;
            v2f a, b;
#pragma unroll
            for (int jj = 0; jj < 2; ++jj) {
                float dx = x0 - xt[p+jj][0];
                float dy = x1 - xt[p+jj][1];
                float dz = x2 - xt[p+jj][2];
                float d2 = dx*dx + dy*dy + dz*dz;
                a[jj] = __expf(-sqrtf(fmaxf(d2, KEPS)) * invs);
                b[jj] = as[p+jj][ln];
            }
            c = wmma4(a, b, c);
        }
        __syncthreads();
    }
#pragma unroll
    for (int r = 0; r < 8; ++r) {
        int row = rowbase + r + 8*half;
        if (row < NPTS) h[(size_t)row * HTOT + l*H + ln] = c[r];
    }
}

// ------------------------------- MLP layer 1 (112 -> 40) with WMMA f32
__global__ void mlp_layer1(const float* __restrict__ h, const float* __restrict__ w1,
                           const float* __restrict__ b1, float* __restrict__ z1) {
    __shared__ float sw[HTOT][48];
    __shared__ float sb[48];
    int tid = threadIdx.x;          // 256
    for (int t = tid; t < HTOT * 40; t += 256) sw[t/40][t%40] = w1[t];
    for (int t = tid; t < HTOT * 8;  t += 256) sw[t/8][40 + t%8] = 0.f;
    if (tid < 48) sb[tid] = (tid < 40) ? b1[tid] : 0.f;
    __syncthreads();
    int wave = tid / 32, lane = tid % 32;
    int half = lane / 16, ln = lane % 16;
    int rowbase = blockIdx.x * 128 + wave * 16;
    int ar = rowbase + ln; if (ar >= NPTS) ar = NPTS - 1;
    const float* hr = h + (size_t)ar * HTOT;
    v8f c0 = {0.f,0.f,0.f,0.f,0.f,0.f,0.f,0.f};
    v8f c1 = c0, c2 = c0;
#pragma unroll
    for (int kk = 0; kk < HTOT / 4; ++kk) {     // 28 K-steps
        int p = kk * 4 + 2 * half;
        v2f a;  a[0] = hr[p]; a[1] = hr[p+1];
        v2f b0; b0[0] = sw[p][ln];      b0[1] = sw[p+1][ln];
        v2f b1v; b1v[0] = sw[p][16+ln]; b1v[1] = sw[p+1][16+ln];
        v2f b2v; b2v[0] = sw[p][32+ln]; b2v[1] = sw[p+1][32+ln];
        c0 = wmma4(a, b0, c0);
        c1 = wmma4(a, b1v, c1);
        c2 = wmma4(a, b2v, c2);
    }
#pragma unroll
    for (int r = 0; r < 8; ++r) {
        int row = rowbase + r + 8*half;
        if (row < NPTS) {
            z1[(size_t)row*40 + ln]      = fmaxf(c0[r] + sb[ln], 0.f);
            z1[(size_t)row*40 + 16 + ln] = fmaxf(c1[r] + sb[16+ln], 0.f);
            if (ln < 8) z1[(size_t)row*40 + 32 + ln] = fmaxf(c2[r] + sb[32+ln], 0.f);
        }
    }
}

// ------------------------------- MLP layers 2..5, register resident
__global__ void mlp_tail(const float* __restrict__ z1,
                         const float* __restrict__ w2, const float* __restrict__ b2,
                         const float* __restrict__ w3, const float* __restrict__ b3,
                         const float* __restrict__ w4, const float* __restrict__ b4,
                         const float* __restrict__ w5, const float* __restrict__ b5,
                         float* __restrict__ out) {
    __shared__ float s2[1600], s3[1600], s4[1600];
    __shared__ float t2[40], t3[40], t4[40], s5[40];
    int tid = threadIdx.x;          // 256
    for (int t = tid; t < 1600; t += 256) { s2[t] = w2[t]; s3[t] = w3[t]; s4[t] = w4[t]; }
    if (tid < 40) { t2[tid] = b2[tid]; t3[tid] = b3[tid]; t4[tid] = b4[tid]; s5[tid] = w5[tid]; }
    __syncthreads();
    int row = blockIdx.x * 256 + tid;
    if (row >= NPTS) return;
    float a[40], b[40];
#pragma unroll
    for (int k = 0; k < 40; ++k) a[k] = z1[(size_t)row*40 + k];
    for (int j = 0; j < 40; ++j) {
        float acc = t2[j];
#pragma unroll
        for (int k = 0; k < 40; ++k) acc += a[k] * s2[k*40 + j];
        b[j] = fmaxf(acc, 0.f);
    }
    for (int j = 0; j < 40; ++j) {
        float acc = t3[j];
#pragma unroll
        for (int k = 0; k < 40; ++k) acc += b[k] * s3[k*40 + j];
        a[j] = fmaxf(acc, 0.f);
    }
    for (int j = 0; j < 40; ++j) {
        float acc = t4[j];
#pragma unroll
        for (int k = 0; k < 40; ++k) acc += a[k] * s4[k*40 + j];
        b[j] = fmaxf(acc, 0.f);
    }
    float o = b5[0];
#pragma unroll
    for (int k = 0; k < 40; ++k) o += b[k] * s5[k];
    out[row] = o;
}

// ------------------------------------------------------------------- driver
extern "C" void kernel_launch(void* const* d_in, const int* in_sizes, int n_in,
                              void* d_out, int out_size, void* d_ws, size_t ws_size,
                              hipStream_t stream) {
    (void)in_sizes; (void)n_in; (void)out_size; (void)ws_size;
    const float* x     = (const float*)d_in[0];
    const float* tp    = (const float*)d_in[1];
    const float* W_emb = (const float*)d_in[2];
    const float* alpha = (const float*)d_in[3];
    const float* sigma = (const float*)d_in[4];
    const float* w1 = (const float*)d_in[5];  const float* b1 = (const float*)d_in[6];
    const float* w2 = (const float*)d_in[7];  const float* b2 = (const float*)d_in[8];
    const float* w3 = (const float*)d_in[9];  const float* b3 = (const float*)d_in[10];
    const float* w4 = (const float*)d_in[11]; const float* b4 = (const float*)d_in[12];
    const float* w5 = (const float*)d_in[13]; const float* b5 = (const float*)d_in[14];
    float* out = (float*)d_out;

    float* ws = (float*)d_ws;
    float* A  = ws;                               // 2048*2048
    float* Y  = A  + (size_t)NN * NN;             // 2048*16
    float* AC = Y  + (size_t)NN * H;              // 2048*16
    float* Hh = AC + (size_t)NN * H;              // 20000*112
    float* Z1 = Hh + (size_t)NPTS * HTOT;         // 20000*40

    for (int l = 0; l < NLVL; ++l) {
        build_kxx<<<(NN * NN) / 256, 256, 0, stream>>>(A, tp, sigma, alpha, l);
        for (int kb = 0; kb < NN / CB; ++kb) {
            int k = kb * CB;
            chol_potrf<<<1, CB, 0, stream>>>(A, k);
            int T = NN - k - CB;
            if (T > 0) {
                chol_trsm<<<T / CB, CB, 0, stream>>>(A, k);
                int nb = T / CB;
                chol_syrk<<<nb * nb, 256, 0, stream>>>(A, k);
            }
        }
        solve_fwd<<<1, 1024, 0, stream>>>(A, W_emb, Y, l);
        solve_bwd<<<1, 1024, 0, stream>>>(A, Y, AC);
        ktx_gemm<<<(NPTS + 127) / 128, 256, 0, stream>>>(x, tp, sigma, AC, Hh, l);
    }
    mlp_layer1<<<(NPTS + 127) / 128, 256, 0, stream>>>(Hh, w1, b1, Z1);
    mlp_tail<<<(NPTS + 255) / 256, 256, 0, stream>>>(Z1, w2, b2, w3, b3, w4, b4, w5, b5, out);
}